// GraphAttentionLayer_11716670784207
// MI455X (gfx1250) — compile-verified
//
#include <hip/hip_runtime.h>
#include <hip/hip_bf16.h>

typedef __attribute__((ext_vector_type(2))) float v2f;
typedef __attribute__((ext_vector_type(8))) float v8f;
typedef int v4i __attribute__((vector_size(4 * sizeof(int))));

#define NN      8192
#define FIN     512
#define FOUT    256
#define GALPHA  0.2f
#define NEG_PAD (-2147483648.0f)   // -2^31 as f32, matches reference mask value

#define AS1 __attribute__((address_space(1)))
#define AS3 __attribute__((address_space(3)))

#if defined(__has_builtin)
#  if __has_builtin(__builtin_amdgcn_global_store_async_from_lds_b128)
#    define HAVE_ASYNC_STORE 1
#  endif
#  if __has_builtin(__builtin_amdgcn_s_wait_asynccnt)
#    define HAVE_WAIT_ASYNC 1
#  endif
#endif

// ---------------------------------------------------------------------------
// Kernel 0: wa1 = W @ a[:256], wa2 = W @ a[256:]   (512 outputs each, tiny)
// ---------------------------------------------------------------------------
__global__ void __launch_bounds__(256)
gat_wa_kernel(const float* __restrict__ W, const float* __restrict__ a,
              float* __restrict__ wa) {
  int r = blockIdx.x * blockDim.x + threadIdx.x;   // 0..511
  if (r < FIN) {
    const float* wr = W + (size_t)r * FOUT;
    float acc1 = 0.f, acc2 = 0.f;
#pragma unroll 4
    for (int c = 0; c < FOUT; ++c) {
      float w = wr[c];
      acc1 = fmaf(w, a[c], acc1);
      acc2 = fmaf(w, a[FOUT + c], acc2);
    }
    wa[r]       = acc1;   // wa1
    wa[FIN + r] = acc2;   // wa2
  }
}

// ---------------------------------------------------------------------------
// Kernel 1: s1 = h @ wa1, s2 = h @ wa2 via V_WMMA_F32_16X16X4_F32.
// One wave per 16-row tile of h; B is the 512x16 matrix with col0 = wa1,
// col1 = wa2, other cols zero. B fragment is built BRANCHLESSLY (pointer
// select + 0/1 mask) so EXEC stays all-ones and the K-loop has no saveexec.
//
// A (16x4 f32): lanes 0-15 -> M=lane, K={0,1}; lanes 16-31 -> M=lane-16, K={2,3}.
// B (4x16 f32): mirrors A with lane = N.
// D (16x16 f32): VGPR r -> M=r (lanes 0-15, N=lane) / M=r+8 (lanes 16-31).
// ---------------------------------------------------------------------------
__global__ void __launch_bounds__(32)
gat_scores_wmma(const float* __restrict__ h, const float* __restrict__ wa,
                float* __restrict__ s1, float* __restrict__ s2) {
  const int lane     = threadIdx.x;        // 0..31
  const int m        = lane & 15;          // A-row / B-col index
  const int koff     = (lane >> 4) << 1;   // 0 for lanes 0-15, 2 for 16-31
  const int row_base = blockIdx.x * 16;

  const float* __restrict__ hrow = h + (size_t)(row_base + m) * FIN;
  // Branchless B operand source: lane m==1 reads wa2, everything else wa1;
  // lanes m>=2 multiply by 0 so their column contributes nothing.
  const float* __restrict__ bsrc = (m == 1) ? (wa + FIN) : wa;
  const float bscale = (m <= 1) ? 1.0f : 0.0f;

  v8f c = {};
#pragma unroll 8
  for (int k = 0; k < FIN; k += 4) {
    v2f aF, bF;
    aF.x = hrow[k + koff];
    aF.y = hrow[k + koff + 1];
    bF.x = bscale * bsrc[k + koff];
    bF.y = bscale * bsrc[k + koff + 1];
    c = __builtin_amdgcn_wmma_f32_16x16x4_f32(
        /*neg_a=*/false, aF, /*neg_b=*/false, bF,
        /*c_mod=*/(short)0, c, /*reuse_a=*/false, /*reuse_b=*/false);
  }

  // Scatter column 0 -> s1, column 1 -> s2 (lanes 0,16 and 1,17 hold them).
  if (m <= 1) {
    float* dst = (m == 0) ? s1 : s2;
    int base = row_base + ((lane < 16) ? 0 : 8);
#pragma unroll
    for (int r = 0; r < 8; ++r) dst[base + r] = c[r];
  }
}

// ---------------------------------------------------------------------------
// Kernel 2: masked row softmax. One block per row; adj row read ONCE from HBM
// (prefetched ahead), masked scores staged in 32KB LDS, exp/normalize from
// LDS. Write-out uses GLOBAL_STORE_ASYNC_FROM_LDS_B128 when available: the
// finished row streams LDS -> HBM without VGPR staging (ASYNCcnt tracked).
// HBM traffic per row = 32KB read + 32KB write (bandwidth-optimal).
// ---------------------------------------------------------------------------
__global__ void __launch_bounds__(256)
gat_softmax(const int* __restrict__ adj, const float* __restrict__ s1,
            const float* __restrict__ s2, float* __restrict__ out) {
  __shared__ float rowbuf[NN];     // 32 KB of the 320 KB WGP LDS
  __shared__ float red[256];

  const int    tid   = threadIdx.x;
  const int    row   = blockIdx.x;
  const size_t rbase = (size_t)row * NN;
  const float  s1i   = s1[row];

  // Pass 1: mask + leaky-relu, track max, stage into LDS.
  float mx = NEG_PAD;
  for (int j = tid; j < NN; j += 256) {
    __builtin_prefetch(&adj[rbase + j + 2048], 0, 1);  // speculative stream-ahead
    int   av = adj[rbase + j];
    float x  = s1i + s2[j];
    float e  = (x > 0.f) ? x : GALPHA * x;
    float v  = (av > 0) ? e : NEG_PAD;
    rowbuf[j] = v;
    mx = fmaxf(mx, v);
  }
  red[tid] = mx;
  __syncthreads();
  for (int s = 128; s > 0; s >>= 1) {
    if (tid < s) red[tid] = fmaxf(red[tid], red[tid + s]);
    __syncthreads();
  }
  const float rowmax = red[0];
  __syncthreads();

  // Pass 2: exponentiate (LDS only), accumulate sum.
  float sum = 0.f;
  for (int j = tid; j < NN; j += 256) {
    float p = __expf(rowbuf[j] - rowmax);
    rowbuf[j] = p;
    sum += p;
  }
  red[tid] = sum;
  __syncthreads();
  for (int s = 128; s > 0; s >>= 1) {
    if (tid < s) red[tid] += red[tid + s];
    __syncthreads();
  }
  const float inv = 1.0f / red[0];
  __syncthreads();

#if defined(HAVE_ASYNC_STORE)
  // Pass 3a: normalize in LDS, then async-stream the whole row LDS -> HBM.
  for (int j = tid; j < NN; j += 256) rowbuf[j] *= inv;
  __syncthreads();
#pragma unroll
  for (int j4 = tid * 4; j4 < NN; j4 += 256 * 4) {
    AS1 v4i* gp = (AS1 v4i*)(unsigned long long)(out + rbase + j4);
    AS3 v4i* lp = (AS3 v4i*)(unsigned int)(unsigned long long)&rowbuf[j4];
    __builtin_amdgcn_global_store_async_from_lds_b128(gp, lp, 0, 0);
  }
#  if defined(HAVE_WAIT_ASYNC)
  __builtin_amdgcn_s_wait_asynccnt(0);
#  endif
#else
  // Pass 3b: normalize and stream out through VGPRs.
  for (int j = tid; j < NN; j += 256) {
    out[rbase + j] = rowbuf[j] * inv;
  }
#endif
}

// ---------------------------------------------------------------------------
extern "C" void kernel_launch(void* const* d_in, const int* in_sizes, int n_in,
                              void* d_out, int out_size, void* d_ws, size_t ws_size,
                              hipStream_t stream) {
  const float* h   = (const float*)d_in[0];   // (8192, 512)
  const int*   adj = (const int*)  d_in[1];   // (8192, 8192)
  const float* W   = (const float*)d_in[2];   // (512, 256)
  const float* a   = (const float*)d_in[3];   // (512, 1)
  float*       out = (float*)d_out;           // (8192, 8192)

  float* ws = (float*)d_ws;
  float* wa = ws;                // 1024 floats: wa1 | wa2
  float* s1 = ws + 1024;         // 8192 floats
  float* s2 = ws + 1024 + NN;    // 8192 floats

  gat_wa_kernel   <<<2,    256, 0, stream>>>(W, a, wa);
  gat_scores_wmma <<<NN/16, 32, 0, stream>>>(h, wa, s1, s2);
  gat_softmax     <<<NN,   256, 0, stream>>>(adj, s1, s2, out);
}